// MyLossFunc_78975858639700
// MI455X (gfx1250) — compile-verified
//
#include <hip/hip_runtime.h>
#include <hip/hip_bf16.h>
#include <stdint.h>

// Problem constants (match reference)
#define B_  8
#define C_  16
#define N_  512
#define NN_ (512 * 512)
#define E_  1024

typedef __attribute__((ext_vector_type(2))) float v2f;
typedef __attribute__((ext_vector_type(4))) float v4f;
typedef __attribute__((ext_vector_type(8))) float v8f;

// ---------------------------------------------------------------------------
// Wave32 full-precision sum reduction using the f32 WMMA pipe:
//   A (16x4 f32): VGPR0 = per-lane value, VGPR1 = 0   -> each lane value
//   appears exactly once in A. B = all-ones (4x16). D = A x B: every row of D
//   is constant = sum of the two lane values mapped to that row. Summing the
//   8 D registers in-lane gives the half-wave partition sum; one xor-16
//   shuffle-add completes the full 32-lane sum (broadcast to all lanes).
// Fixed dataflow -> bitwise deterministic. Requires EXEC == all ones.
// ---------------------------------------------------------------------------
__device__ __forceinline__ float wave_reduce_sum_wmma(float s) {
  v2f a;   a[0] = s;    a[1] = 0.0f;
  v2f bo;  bo[0] = 1.0f; bo[1] = 1.0f;
  v8f cz = {0.f, 0.f, 0.f, 0.f, 0.f, 0.f, 0.f, 0.f};
  v8f d = __builtin_amdgcn_wmma_f32_16x16x4_f32(
      /*neg_a=*/false, a, /*neg_b=*/false, bo,
      /*c_mod=*/(short)0, cz, /*reuse_a=*/false, /*reuse_b=*/false);
  float r = d[0] + d[1] + d[2] + d[3] + d[4] + d[5] + d[6] + d[7];
  r += __shfl_xor(r, 16, 32);   // combine the two 16-lane halves
  return r;
}

// ---------------------------------------------------------------------------
// Stage 0: zero the u8 label grid (2 MiB) as 32-bit stores.
// ---------------------------------------------------------------------------
__global__ void __launch_bounds__(256) zero_labels_kernel(uint32_t* __restrict__ l32) {
  l32[(size_t)blockIdx.x * 256 + threadIdx.x] = 0u;
}

// ---------------------------------------------------------------------------
// Stage 1: scatter gold edge labels. One thread per batch, sequential over
// edges -> deterministic last-write-wins for duplicate indices.
// ---------------------------------------------------------------------------
__global__ void scatter_edges_kernel(const int* __restrict__ edge_index,
                                     const int* __restrict__ edge_attr,
                                     unsigned char* __restrict__ labels) {
  int b = threadIdx.x;
  if (b >= B_) return;
  for (int e = 0; e < E_; ++e) {
    int row = edge_index[((size_t)b * E_ + e) * 2 + 0];
    int col = edge_index[((size_t)b * E_ + e) * 2 + 1];
    int a   = edge_attr[(size_t)b * E_ + e];
    labels[(size_t)b * NN_ + (size_t)row * N_ + col] = (unsigned char)a;
  }
}

// ---------------------------------------------------------------------------
// Stage 2: streaming CE. Grid = 2048 blocks x 256 threads, thread handles 4
// consecutive j positions -> 16 coalesced non-temporal b128 loads per thread.
// Block writes one deterministic (sum, count) partial per block.
// ---------------------------------------------------------------------------
__global__ void __launch_bounds__(256) ce_main_kernel(
    const float* __restrict__ adj, const int* __restrict__ mask,
    const unsigned char* __restrict__ labels, float2* __restrict__ partials) {
  const int blocksPerBatch = NN_ / 1024;                       // 256
  const int b   = blockIdx.x / blocksPerBatch;
  const int pos = (blockIdx.x % blocksPerBatch) * 1024 + (int)threadIdx.x * 4;

  const float* abase = adj + (size_t)b * C_ * NN_ + pos;

  float x[C_][4];
#pragma unroll
  for (int c = 0; c < C_; ++c) {
    v4f t = __builtin_nontemporal_load((const v4f*)(abase + (size_t)c * NN_));
    x[c][0] = t[0]; x[c][1] = t[1]; x[c][2] = t[2]; x[c][3] = t[3];
  }
  uchar4 lab = *(const uchar4*)(labels + (size_t)b * NN_ + pos);
  int4   mk  = *(const int4*)(mask + (size_t)b * NN_ + pos);

  const int lv[4] = {lab.x, lab.y, lab.z, lab.w};
  const int mv[4] = {mk.x, mk.y, mk.z, mk.w};

  float s = 0.f, cnt = 0.f;
#pragma unroll
  for (int k = 0; k < 4; ++k) {
    float m = x[0][k];
#pragma unroll
    for (int c = 1; c < C_; ++c) m = fmaxf(m, x[c][k]);
    float sum = 0.f;
    float xl  = x[0][k];
#pragma unroll
    for (int c = 0; c < C_; ++c) {
      sum += __expf(x[c][k] - m);
      if (c > 0) xl = (lv[k] == c) ? x[c][k] : xl;   // cndmask chain, no dyn index
    }
    float lse   = m + __logf(sum);
    float valid = mv[k] ? 1.0f : 0.0f;
    s   += (lse - xl) * valid;
    cnt += valid;
  }

  // Per-wave WMMA reductions (all 256 threads alive -> EXEC all ones here).
  float rs = wave_reduce_sum_wmma(s);
  float rc = wave_reduce_sum_wmma(cnt);

  __shared__ float2 wpart[8];
  const int wave = threadIdx.x >> 5;
  const int lane = threadIdx.x & 31;
  if (lane == 0) wpart[wave] = make_float2(rs, rc);
  __syncthreads();
  if (threadIdx.x == 0) {
    float S = 0.f, Cn = 0.f;
    for (int w = 0; w < 8; ++w) { S += wpart[w].x; Cn += wpart[w].y; }
    partials[blockIdx.x] = make_float2(S, Cn);
  }
}

// ---------------------------------------------------------------------------
// Stage 3: final deterministic reduce. One block of 256; warp b reduces the
// 256 partials of batch b (fixed order), WMMA wave-reduces, computes the
// per-sample mean, thread 0 averages over batches.
// ---------------------------------------------------------------------------
__global__ void __launch_bounds__(256) ce_final_kernel(
    const float2* __restrict__ partials, float* __restrict__ out) {
  const int wave = threadIdx.x >> 5;   // batch id (B_ == 8 waves)
  const int lane = threadIdx.x & 31;

  float s = 0.f, c = 0.f;
#pragma unroll
  for (int k = 0; k < 8; ++k) {
    float2 p = partials[wave * 256 + k * 32 + lane];
    s += p.x; c += p.y;
  }
  float rs = wave_reduce_sum_wmma(s);
  float rc = wave_reduce_sum_wmma(c);

  __shared__ float per_sample[B_];
  if (lane == 0) per_sample[wave] = rs / fmaxf(rc, 1.0f);
  __syncthreads();
  if (threadIdx.x == 0) {
    float t = 0.f;
    for (int b = 0; b < B_; ++b) t += per_sample[b];
    out[0] = t * (1.0f / (float)B_);
  }
}

// ---------------------------------------------------------------------------
extern "C" void kernel_launch(void* const* d_in, const int* in_sizes, int n_in,
                              void* d_out, int out_size, void* d_ws, size_t ws_size,
                              hipStream_t stream) {
  const float* adj        = (const float*)d_in[0];            // [B,C,N,N] f32
  const int*   mask       = (const int*)d_in[1];              // [B,N,N] bool->i32
  const int*   edge_index = (const int*)d_in[2];              // [B,E,2] i32
  const int*   edge_attr  = (const int*)d_in[3];              // [B,E] i32
  float*       out        = (float*)d_out;                    // scalar f32

  // Workspace layout: [0, 2 MiB) u8 label grid; then 2048 float2 partials.
  unsigned char* labels   = (unsigned char*)d_ws;
  float2*        partials = (float2*)((char*)d_ws + (size_t)B_ * NN_);

  zero_labels_kernel<<<(B_ * NN_ / 4) / 256, 256, 0, stream>>>((uint32_t*)labels);
  scatter_edges_kernel<<<1, 32, 0, stream>>>(edge_index, edge_attr, labels);
  ce_main_kernel<<<B_ * NN_ / 1024, 256, 0, stream>>>(adj, mask, labels, partials);
  ce_final_kernel<<<1, 256, 0, stream>>>(partials, out);
}